// GCN_63299228009239
// MI455X (gfx1250) — compile-verified
//
#include <hip/hip_runtime.h>
#include <hip/hip_bf16.h>

typedef __attribute__((ext_vector_type(2))) float v2f;
typedef __attribute__((ext_vector_type(8))) float v8f;

#define DIN 128
#define DOUT 128

// ---------------------------------------------------------------------------
// Kernel 1: agg = X (self contribution), deg = 0.  (workspace is poisoned to
// 0xAA by the harness, so this must run every call.)
// ---------------------------------------------------------------------------
__global__ void gcn_init(const float* __restrict__ X, float* __restrict__ agg,
                         float* __restrict__ deg, int n_nodes) {
    int i = blockIdx.x * blockDim.x + threadIdx.x;
    int total = n_nodes * DIN;
    if (i < total) agg[i] = X[i];
    if (i < n_nodes) deg[i] = 0.0f;
}

// ---------------------------------------------------------------------------
// Kernel 2: degree counting.  One thread per edge, two f32 atomics.
// ---------------------------------------------------------------------------
__global__ void gcn_degree(const long long* __restrict__ edges, float* __restrict__ deg,
                           int n_edges) {
    int i = blockIdx.x * blockDim.x + threadIdx.x;
    if (i >= n_edges) return;
    long long s = edges[i];
    long long d = edges[(long long)n_edges + i];
    unsafeAtomicAdd(&deg[s], 1.0f);
    unsafeAtomicAdd(&deg[d], 1.0f);
}

// ---------------------------------------------------------------------------
// Kernel 3: neighbor scatter-add.  One wave32 per edge; each lane owns a
// float4 slice (32 lanes * 4 = 128 features).  Both directions handled by the
// same wave: agg[dst] += X[src], agg[src] += X[dst].
// All traffic is L2-resident (X and agg are ~51MB each, L2 is 192MB).
// ---------------------------------------------------------------------------
__global__ void gcn_scatter(const long long* __restrict__ edges,
                            const float* __restrict__ X,
                            float* __restrict__ agg, int n_edges) {
    int wave = threadIdx.x >> 5;
    int lane = threadIdx.x & 31;
    long long e = (long long)blockIdx.x * 8 + wave;
    if (e >= n_edges) return;

    long long s = edges[e];
    long long d = edges[(long long)n_edges + e];

    const float4 xs = ((const float4*)(X + s * DIN))[lane];
    const float4 xd = ((const float4*)(X + d * DIN))[lane];

    float* ad = agg + d * DIN + lane * 4;
    float* as = agg + s * DIN + lane * 4;

    unsafeAtomicAdd(ad + 0, xs.x);
    unsafeAtomicAdd(ad + 1, xs.y);
    unsafeAtomicAdd(ad + 2, xs.z);
    unsafeAtomicAdd(ad + 3, xs.w);
    unsafeAtomicAdd(as + 0, xd.x);
    unsafeAtomicAdd(as + 1, xd.y);
    unsafeAtomicAdd(as + 2, xd.z);
    unsafeAtomicAdd(as + 3, xd.w);
}

// ---------------------------------------------------------------------------
// Kernel 4: out = relu( (agg/deg) @ W ) using V_WMMA_F32_16X16X4_F32.
//
// Block = 256 threads = 8 waves.  Each wave computes 16 rows x 128 cols:
// 8 v8f accumulator tiles, K stepping by 4, 8 WMMA per step.
//
// W is staged in LDS as an XOR-swizzled transpose:
//     Ws[n*128 + (k ^ ((n&31)<<1))] = W[k*128 + n]
// so a B fragment {W[k][n], W[k+1][n]} is ONE aligned ds_load_b64 into a
// consecutive even VGPR pair (swizzle term is even, k is even), and the
// access is bank-conflict-free within each half-wave (cross-half lanes hit the same
// k^s offset when ncol ^ ncol' == 1, since k0+2 == k0 ^ 2).
//
// ISA fragment layouts (CDNA5 ISA):
//   A 16x4 f32 : lanes 0-15 hold {K=k0,k0+1}, lanes 16-31 hold {K=k0+2,k0+3}
//                -> per-lane contiguous global_load_b64 at H[row][k0+koff]
//   B 4x16 f32 : V0: lanes0-15 K=k0 / lanes16-31 K=k0+2; V1: k0+1 / k0+3,
//                N = lane&15  -> one ds_load_b64 from the swizzled transpose
//   C/D 16x16  : VGPR r -> M = r (lanes 0-15) or r+8 (lanes 16-31), N = lane&15
// ---------------------------------------------------------------------------
__global__ void gcn_wmma_gemm(const float* __restrict__ agg,
                              const float* __restrict__ deg,
                              const float* __restrict__ W,
                              float* __restrict__ out, int n_rows) {
    __shared__ float Ws[DIN * DOUT];  // 64 KB, swizzled transpose

    for (int i = threadIdx.x; i < DIN * DOUT; i += blockDim.x) {
        int k = i >> 7;
        int n = i & 127;
        Ws[n * DIN + (k ^ ((n & 31) << 1))] = W[i];
    }
    __syncthreads();

    int wave = threadIdx.x >> 5;
    int lane = threadIdx.x & 31;
    int rowBase = blockIdx.x * 128 + wave * 16;
    if (rowBase >= n_rows) return;  // wave-uniform: EXEC stays all-ones

    int mrow = rowBase + (lane & 15);      // A/D row this lane feeds
    int koff = (lane >> 4) << 1;           // 0 for lanes 0-15, 2 for 16-31
    int ncol = lane & 15;                  // within-tile output column
    float rdeg = 1.0f / deg[mrow];         // fold normalization into A load
    const float* Arow = agg + (long long)mrow * DIN;

    int s0 = ncol << 1;                    // swizzle for even tiles (n&31 = ncol)
    int s1 = (ncol + 16) << 1;             // swizzle for odd tiles (n&31 = ncol+16)
    const float* WsCol = Ws + ncol * DIN;  // per-lane column base; +t*16*DIN is
                                           // a compile-time DS immediate offset

    v8f acc[8] = {};

    for (int k0 = 0; k0 < DIN; k0 += 4) {
        int k = k0 + koff;
        v2f a = *(const v2f*)(Arow + k);
        a *= rdeg;
        int e0 = k ^ s0;                   // swizzled k-offset, even tiles
        int e1 = k ^ s1;                   // swizzled k-offset, odd tiles
#pragma unroll
        for (int t = 0; t < 8; ++t) {
            v2f b = *(const v2f*)(WsCol + t * 16 * DIN + ((t & 1) ? e1 : e0));
            acc[t] = __builtin_amdgcn_wmma_f32_16x16x4_f32(
                false, a, false, b, (short)0, acc[t], false, false);
        }
    }

    int rAdd = (lane >> 4) * 8;  // upper half-wave owns M = r+8
#pragma unroll
    for (int t = 0; t < 8; ++t) {
#pragma unroll
        for (int r = 0; r < 8; ++r) {
            float v = acc[t][r];
            out[(long long)(rowBase + r + rAdd) * DOUT + t * 16 + ncol] =
                fmaxf(v, 0.0f);
        }
    }
}

// ---------------------------------------------------------------------------
extern "C" void kernel_launch(void* const* d_in, const int* in_sizes, int n_in,
                              void* d_out, int out_size, void* d_ws, size_t ws_size,
                              hipStream_t stream) {
    const float* X = (const float*)d_in[0];
    const float* W = (const float*)d_in[1];
    const long long* edges = (const long long*)d_in[2];
    float* out = (float*)d_out;

    const int n_nodes = in_sizes[0] / DIN;   // 100000
    const int n_edges = in_sizes[2] / 2;     // 1600000

    float* agg = (float*)d_ws;                       // n_nodes * 128 floats
    float* deg = agg + (size_t)n_nodes * DIN;        // n_nodes floats

    {   // init agg = X, deg = 0
        int total = n_nodes * DIN;
        gcn_init<<<(total + 255) / 256, 256, 0, stream>>>(X, agg, deg, n_nodes);
    }
    {   // degree atomics
        gcn_degree<<<(n_edges + 255) / 256, 256, 0, stream>>>(edges, deg, n_edges);
    }
    {   // feature scatter: one wave per edge, 8 edges per block
        gcn_scatter<<<(n_edges + 7) / 8, 256, 0, stream>>>(edges, X, agg, n_edges);
    }
    {   // normalize + GEMM + relu, 128 rows per block
        gcn_wmma_gemm<<<(n_nodes + 127) / 128, 256, 0, stream>>>(agg, deg, W, out, n_nodes);
    }
}